// MPModule_12214886990221
// MI455X (gfx1250) — compile-verified
//
#include <hip/hip_runtime.h>

// MI455X / gfx1250, wave32. fp32 WMMA (V_WMMA_F32_16X16X4_F32) everywhere:
// after reassociating sim@X = e(e^T X)/D the problem is ~7.6 GFLOP / ~80 MB,
// i.e. near the bandwidth roofline even at f32 matrix throughput.
// Round 2: batched A-fragment loads (no per-load s_wait_loadcnt 0) and
// 4 independent WMMA accumulator chains per wave to break the C->D RAW chain.

typedef __attribute__((ext_vector_type(2))) float v2f;
typedef __attribute__((ext_vector_type(8))) float v8f;

#define DD     256
#define NEDGE  8192
#define NM     (NEDGE * DD)
#define KSPLIT 8
#define KCH    (NEDGE / KSPLIT)

__device__ __forceinline__ v8f wmma4(v2f a, v2f b, v8f c) {
    // D = A(16x4,f32) * B(4x16,f32) + C(16x16,f32)
    return __builtin_amdgcn_wmma_f32_16x16x4_f32(false, a, false, b,
                                                 (short)0, c, false, false);
}

// ---------------------------------------------------------------------------
// Kernel 1: E = relu(X @ W + b), for (W1,b1)->E1 and (W2,b2)->E2 (blockIdx.z)
// Block = 8 waves. Wave owns a 16-col strip x 128 rows; the 64 B-fragments
// (full K=256) are hoisted into VGPRs. Two M-tiles in flight, each with two
// split accumulators -> 4 independent WMMA chains; A loads batched in 8s.
// ---------------------------------------------------------------------------
__global__ __launch_bounds__(256) void k_linear_relu(
    const float* __restrict__ X,
    const float* __restrict__ W1, const float* __restrict__ b1,
    const float* __restrict__ W2, const float* __restrict__ b2,
    float* __restrict__ E1, float* __restrict__ E2)
{
    const int wave = threadIdx.x >> 5;
    const int lane = threadIdx.x & 31;
    const int lm   = lane & 15;
    const int lk   = (lane >> 4) << 1;

    const float* W  = blockIdx.z ? W2 : W1;
    const float* bb = blockIdx.z ? b2 : b1;
    float*       E  = blockIdx.z ? E2 : E1;

    const int n0      = blockIdx.y * 128 + wave * 16;
    const int rowBase = blockIdx.x * 128;

    v2f bf[DD / 4];
#pragma unroll
    for (int ks = 0; ks < DD / 4; ++ks) {
        const int k = 4 * ks + lk;
        v2f b;
        b.x = W[k * DD + n0 + lm];
        b.y = W[(k + 1) * DD + n0 + lm];
        bf[ks] = b;
    }
    const float bn = bb[n0 + lm];

    for (int mt = 0; mt < 8; mt += 2) {
        const int m0 = rowBase + mt * 16;
        const float* x0 = X + (m0 + lm) * DD + lk;
        const float* x1 = x0 + 16 * DD;
        if (mt < 6) {  // warm L0/L2 for the next M-tile pair (global_prefetch_b8)
            __builtin_prefetch(x0 + 32 * DD, 0, 3);
            __builtin_prefetch(x1 + 32 * DD, 0, 3);
        }
        v8f c0a = {}, c0b = {}, c1a = {}, c1b = {};
#pragma unroll
        for (int kb = 0; kb < DD / 4; kb += 8) {
            v2f a0[8], a1[8];
#pragma unroll
            for (int j = 0; j < 8; ++j) a0[j] = *(const v2f*)(x0 + 4 * (kb + j));
#pragma unroll
            for (int j = 0; j < 8; ++j) a1[j] = *(const v2f*)(x1 + 4 * (kb + j));
#pragma unroll
            for (int j = 0; j < 8; j += 2) {
                c0a = wmma4(a0[j],     bf[kb + j],     c0a);
                c1a = wmma4(a1[j],     bf[kb + j],     c1a);
                c0b = wmma4(a0[j + 1], bf[kb + j + 1], c0b);
                c1b = wmma4(a1[j + 1], bf[kb + j + 1], c1b);
            }
        }
        const v8f c0 = c0a + c0b;
        const v8f c1 = c1a + c1b;
        const int rb = ((lane >> 4) << 3);
#pragma unroll
        for (int r = 0; r < 8; ++r) {
            float v0 = c0[r] + bn;
            float v1 = c1[r] + bn;
            E[(m0 + rb + r) * DD + n0 + lm]      = v0 > 0.f ? v0 : 0.f;
            E[(m0 + 16 + rb + r) * DD + n0 + lm] = v1 > 0.f ? v1 : 0.f;
        }
    }
}

// ---------------------------------------------------------------------------
// Kernel 2: split-K Gram partials  P[z][s] = E_z[kchunk]^T @ X[kchunk]
// Wave owns a 16x64 tile (4 N-tiles share one A fragment); two k-steps of
// fragments (20 b32 loads) batched ahead of 8 WMMAs on 4 chains.
// ---------------------------------------------------------------------------
__global__ __launch_bounds__(256) void k_gram_partial(
    const float* __restrict__ X,
    const float* __restrict__ E1, const float* __restrict__ E2,
    float* __restrict__ P)          // [2][KSPLIT][DD][DD]
{
    const int wave = threadIdx.x >> 5;
    const int lane = threadIdx.x & 31;
    const int lm   = lane & 15;
    const int lk   = (lane >> 4) << 1;

    const int strip = blockIdx.x * 8 + wave;   // 0..63
    const int m0    = (strip >> 2) * 16;       // E-column tile
    const int n0    = (strip & 3) * 64;        // X-column strip (4 tiles)
    const float* E  = blockIdx.z ? E2 : E1;
    const int kbeg  = blockIdx.y * KCH;

    v8f c[4] = {};
    for (int k = kbeg; k < kbeg + KCH; k += 8) {
        v2f a[2], b[2][4];
#pragma unroll
        for (int t = 0; t < 2; ++t) {
            const float* er = E + (k + 4 * t + lk) * DD;
            const float* xr = X + (k + 4 * t + lk) * DD;
            a[t].x = er[m0 + lm];
            a[t].y = er[DD + m0 + lm];
#pragma unroll
            for (int j = 0; j < 4; ++j) {
                b[t][j].x = xr[n0 + 16 * j + lm];
                b[t][j].y = xr[DD + n0 + 16 * j + lm];
            }
        }
#pragma unroll
        for (int t = 0; t < 2; ++t)
#pragma unroll
            for (int j = 0; j < 4; ++j) c[j] = wmma4(a[t], b[t][j], c[j]);
    }

    float* outp = P + (size_t)(blockIdx.z * KSPLIT + blockIdx.y) * DD * DD;
    const int rb = m0 + ((lane >> 4) << 3);
#pragma unroll
    for (int j = 0; j < 4; ++j)
#pragma unroll
        for (int r = 0; r < 8; ++r)
            outp[(rb + r) * DD + n0 + 16 * j + lm] = c[j][r];
}

// ---------------------------------------------------------------------------
// Kernel 3: H_z = (sum_s P[z][s]) / D  @  W3_block(z+1)
// (agg_z @ W3_block == e_z @ H_z by associativity; G itself never stored)
// ---------------------------------------------------------------------------
__global__ __launch_bounds__(256) void k_reduce_H(
    const float* __restrict__ P,    // [2][KSPLIT][DD][DD]
    const float* __restrict__ W3,   // [3*DD][DD]
    float* __restrict__ H)          // [2][DD][DD]
{
    const int wave = threadIdx.x >> 5;
    const int lane = threadIdx.x & 31;
    const int lm   = lane & 15;
    const int lk   = (lane >> 4) << 1;

    const int tile = blockIdx.x * 8 + wave;    // 0..255
    const int m0   = (tile >> 4) * 16;
    const int n0   = (tile & 15) * 16;
    const int z    = blockIdx.y;
    const float* W3b = W3 + (size_t)(z + 1) * DD * DD;
    const float invD = 1.0f / (float)DD;

    v8f c = {};
#pragma unroll 4
    for (int ks = 0; ks < DD / 4; ++ks) {
        const int k = 4 * ks + lk;
        float ax = 0.f, ay = 0.f;
#pragma unroll
        for (int s = 0; s < KSPLIT; ++s) {
            const float* pr = P + ((size_t)(z * KSPLIT + s) * DD + m0 + lm) * DD + k;
            v2f t = *(const v2f*)pr;
            ax += t.x; ay += t.y;
        }
        v2f a; a.x = ax * invD; a.y = ay * invD;
        v2f b;
        b.x = W3b[k * DD + n0 + lm];
        b.y = W3b[(k + 1) * DD + n0 + lm];
        c = wmma4(a, b, c);
    }

    float* out = H + (size_t)z * DD * DD;
    const int rb = m0 + ((lane >> 4) << 3);
#pragma unroll
    for (int r = 0; r < 8; ++r)
        out[(rb + r) * DD + n0 + lm] = c[r];
}

// ---------------------------------------------------------------------------
// Kernel 4: out = relu(X@W3_top + E1@H1 + E2@H2 + b3)
// Wave owns 16 cols x 32 rows; per source, hoist 64 B-frags then run the
// 2 M-tiles with split accumulators (4 chains) and batched A loads.
// ---------------------------------------------------------------------------
__global__ __launch_bounds__(256) void k_final(
    const float* __restrict__ X,
    const float* __restrict__ E1, const float* __restrict__ E2,
    const float* __restrict__ W3, const float* __restrict__ b3,
    const float* __restrict__ H,   // [2][DD][DD]
    float* __restrict__ Y)
{
    const int wave = threadIdx.x >> 5;
    const int lane = threadIdx.x & 31;
    const int lm   = lane & 15;
    const int lk   = (lane >> 4) << 1;

    const int n0 = blockIdx.y * 128 + wave * 16;
    const int m0 = blockIdx.x * 32;

    const float* Asrc[3] = { X, E1, E2 };
    const float* Bsrc[3] = { W3, H, H + (size_t)DD * DD };

    v8f c0a = {}, c0b = {}, c1a = {}, c1b = {};
#pragma unroll
    for (int s = 0; s < 3; ++s) {
        v2f bf[DD / 4];
#pragma unroll
        for (int ks = 0; ks < DD / 4; ++ks) {
            const int k = 4 * ks + lk;
            v2f b;
            b.x = Bsrc[s][k * DD + n0 + lm];
            b.y = Bsrc[s][(k + 1) * DD + n0 + lm];
            bf[ks] = b;
        }
        const float* a0p = Asrc[s] + (m0 + lm) * DD + lk;
        const float* a1p = a0p + 16 * DD;
        if (s < 2) {  // prefetch next source's A stream
            __builtin_prefetch(Asrc[s + 1] + (m0 + lm) * DD + lk, 0, 3);
            __builtin_prefetch(Asrc[s + 1] + (m0 + 16 + lm) * DD + lk, 0, 3);
        }
#pragma unroll
        for (int kb = 0; kb < DD / 4; kb += 8) {
            v2f a0[8], a1[8];
#pragma unroll
            for (int j = 0; j < 8; ++j) a0[j] = *(const v2f*)(a0p + 4 * (kb + j));
#pragma unroll
            for (int j = 0; j < 8; ++j) a1[j] = *(const v2f*)(a1p + 4 * (kb + j));
#pragma unroll
            for (int j = 0; j < 8; j += 2) {
                c0a = wmma4(a0[j],     bf[kb + j],     c0a);
                c1a = wmma4(a1[j],     bf[kb + j],     c1a);
                c0b = wmma4(a0[j + 1], bf[kb + j + 1], c0b);
                c1b = wmma4(a1[j + 1], bf[kb + j + 1], c1b);
            }
        }
    }

    const v8f c0 = c0a + c0b;
    const v8f c1 = c1a + c1b;
    const float bn = b3[n0 + lm];
    const int rb = ((lane >> 4) << 3);
#pragma unroll
    for (int r = 0; r < 8; ++r) {
        float v0 = c0[r] + bn;
        float v1 = c1[r] + bn;
        Y[(m0 + rb + r) * DD + n0 + lm]      = v0 > 0.f ? v0 : 0.f;
        Y[(m0 + 16 + rb + r) * DD + n0 + lm] = v1 > 0.f ? v1 : 0.f;
    }
}

// ---------------------------------------------------------------------------
extern "C" void kernel_launch(void* const* d_in, const int* in_sizes, int n_in,
                              void* d_out, int out_size, void* d_ws, size_t ws_size,
                              hipStream_t stream) {
    // setup_inputs order: 0 edge_pred, 1 edge_corner(i64), 2 all_corners,
    // 3 edge_x, 4 image_x, 5 W1, 6 b1, 7 W2, 8 b2, 9 W3, 10 b3
    const float* edge_x = (const float*)d_in[3];
    const float* W1 = (const float*)d_in[5];
    const float* b1 = (const float*)d_in[6];
    const float* W2 = (const float*)d_in[7];
    const float* b2 = (const float*)d_in[8];
    const float* W3 = (const float*)d_in[9];
    const float* b3 = (const float*)d_in[10];
    float* out = (float*)d_out;

    // workspace layout (floats): E1 | E2 | P[2][KSPLIT][D][D] | H[2][D][D]
    float* ws = (float*)d_ws;
    float* E1 = ws;
    float* E2 = ws + NM;
    float* P  = ws + 2 * (size_t)NM;
    float* H  = P + 2 * (size_t)KSPLIT * DD * DD;

    k_linear_relu <<<dim3(NEDGE / 128, 2, 2), 256, 0, stream>>>(edge_x, W1, b1, W2, b2, E1, E2);
    k_gram_partial<<<dim3(8, KSPLIT, 2),     256, 0, stream>>>(edge_x, E1, E2, P);
    k_reduce_H    <<<dim3(32, 2),            256, 0, stream>>>(P, W3, H);
    k_final       <<<dim3(NEDGE / 32, 2),    256, 0, stream>>>(edge_x, E1, E2, W3, b3, H, out);
}